// NonIsotropicNCA_29274497089742
// MI455X (gfx1250) — compile-verified
//
#include <hip/hip_runtime.h>

typedef __attribute__((ext_vector_type(2))) float v2f;
typedef __attribute__((ext_vector_type(8))) float v8f;

#define XP 29                      // padded extent for x,y,z (25 + 2+2 halo)
#define WPAD 32                    // padded innermost extent (25 + halo + slack)
#define PLANE (XP * XP * XP * WPAD) // 780448 floats per (batch, channel)

// ---- workspace layout (float offsets) ----
#define OFF_XPAD  0
#define SZ_XPAD   (2 * PLANE)              // B=2, Ci=1
#define OFF_HPAD  (SZ_XPAD + 64)           // +64 slack for benign tail over-reads
#define SZ_HCH    (32 * PLANE)             // 2 batches * 16 channels
#define OFF_GPAD  (OFF_HPAD + SZ_HCH + 64)
#define OFF_AWA   (OFF_GPAD + SZ_HCH + 64) // stage-A weights: [500][32][2]
#define SZ_AWA    (500 * 32 * 2)
#define OFF_AW2A  (OFF_AWA + SZ_AWA)       // conv2a weights:  [8000][16][2]
#define SZ_AW2A   (8000 * 16 * 2)
#define OFF_AW2AT (OFF_AW2A + SZ_AW2A)     // conv2a transposed weights

__device__ __forceinline__ v8f wmma4(v2f a, v2f b, v8f c) {
  // D = A(16x4,f32) * B(4x16,f32) + C(16x16,f32)
  return __builtin_amdgcn_wmma_f32_16x16x4_f32(
      false, a, false, b, (short)0, c, false, false);
}

// ---------------- setup kernels ----------------

__global__ void zero_ws_kernel(float4* ws4, int n4) {
  int i = blockIdx.x * blockDim.x + threadIdx.x;
  if (i < n4) ws4[i] = make_float4(0.f, 0.f, 0.f, 0.f);
}

__global__ void pack_x_kernel(const float* __restrict__ x, float* __restrict__ xpad) {
  int t = blockIdx.x * blockDim.x + threadIdx.x;
  if (t >= 2 * 25 * 25 * 25 * 25) return;
  int src = t;
  int w = t % 25; t /= 25;
  int z = t % 25; t /= 25;
  int y = t % 25; t /= 25;
  int xx = t % 25;
  int bi = t / 25;
  int dst = (((bi * XP + xx + 2) * XP + y + 2) * XP + z + 2) * WPAD + (w + 2);
  xpad[dst] = x[src];
}

// AwA[k2][32][2]; rows 0..7 = w0 (3,3,5,5 embedded), 8..15 = w1,
// 16..23 = w0 transposed, 24..31 = w1 transposed. k = (((p*5+q)*5+r)*8)+s.
__global__ void pack_wA_kernel(const float* __restrict__ w0, const float* __restrict__ w1,
                               float* __restrict__ AwA) {
  int t = blockIdx.x * blockDim.x + threadIdx.x;
  if (t >= 500 * 32 * 2) return;
  int v = t & 1;
  int m = (t >> 1) & 31;
  int k2 = t >> 6;
  int k = 2 * k2 + v;
  int s = k & 7; k >>= 3;
  int r = k % 5; k /= 5;
  int q = k % 5;
  int p = k / 5;
  float val = 0.f;
  if (m < 8) {               // w0: offsets (p-2,q-2) from (ka-1,kb-1) -> ka=p-1
    int co = m;
    if (p >= 1 && p <= 3 && q >= 1 && q <= 3 && s < 5)
      val = w0[((co * 3 + (p - 1)) * 3 + (q - 1)) * 25 + r * 5 + s];
  } else if (m < 16) {       // w1 full 5^4
    int co = m - 8;
    if (s < 5) val = w1[((co * 5 + p) * 5 + q) * 25 + r * 5 + s];
  } else if (m < 24) {       // w0 transposed: w'(p,q,r,s) = w0[r-1,s-1,p,q]
    int co = m - 16;
    if (r >= 1 && r <= 3 && s >= 1 && s <= 3)
      val = w0[((co * 3 + (r - 1)) * 3 + (s - 1)) * 25 + p * 5 + q];
  } else {                   // w1 transposed: w'(p,q,r,s) = w1[r,s,p,q]
    int co = m - 24;
    if (s < 5) val = w1[((co * 5 + r) * 5 + s) * 25 + p * 5 + q];
  }
  AwA[t] = val;
}

// Aw2a[k2][16][2] (and transposed variant). k = ((((ci*5+p)*5+q)*5+r)*8)+s.
__global__ void pack_w2a_kernel(const float* __restrict__ w2a,
                                float* __restrict__ Aw, float* __restrict__ AwT) {
  int t = blockIdx.x * blockDim.x + threadIdx.x;
  if (t >= 8000 * 16 * 2) return;
  int v = t & 1;
  int m = (t >> 1) & 15;
  int k2 = t >> 5;
  int k = 2 * k2 + v;
  int s = k & 7; k >>= 3;
  int r = k % 5; k /= 5;
  int q = k % 5; k /= 5;
  int p = k % 5;
  int ci = k / 5;
  float vn = 0.f, vt = 0.f;
  if (s < 5) {
    vn = w2a[(((m * 16 + ci) * 5 + p) * 5 + q) * 25 + r * 5 + s];
    vt = w2a[(((m * 16 + ci) * 5 + r) * 5 + s) * 25 + p * 5 + q];
  }
  Aw[t] = vn;
  AwT[t] = vt;
}

// ---------------- stage A: h = relu(conv{w0,w1}) + relu(conv{w0T,w1T}) ----------------
// One wave = both 16-wide w-tiles at one (b,x,y,z): M=32 via 2 accumulator pairs,
// A fragments shared across the two N-tiles (8 wmma per 8 b64 loads).

__global__ void __launch_bounds__(128)
convA_kernel(const float* __restrict__ xpad, const float* __restrict__ AwA,
             const float* __restrict__ b0, const float* __restrict__ b1,
             float* __restrict__ hpad) {
  int wave = (blockIdx.x * blockDim.x + threadIdx.x) >> 5;
  int lane = threadIdx.x & 31;
  if (wave >= 2 * 25 * 25 * 25) return;
  int t = wave;
  int zo = t % 25; t /= 25;
  int yo = t % 25; t /= 25;
  int xo = t % 25;
  int bi = t / 25;

  const int n = lane & 15, hh = lane >> 4, m = n;
  v8f c00 = {}, c01 = {};   // w-tile 0: normal rows / transposed rows
  v8f c10 = {}, c11 = {};   // w-tile 1

  for (int p = 0; p < 5; ++p)
    for (int q = 0; q < 5; ++q) {
      const float* srcq =
          xpad + ((((bi * XP + xo + p) * XP) + yo + q) * XP + zo) * WPAD +
          n + 2 * hh;
      int base4 = (p * 5 + q) * 5 * 4;
      const float* aw = AwA + (((base4 + hh) * 32) + m) * 2;
#pragma unroll
      for (int r = 0; r < 5; ++r) {
        const float* s0 = srcq + r * WPAD;
        v2f A0 = *(const v2f*)(aw);             // rows 0..15, k2
        v2f A1 = *(const v2f*)(aw + 32);        // rows 16..31 (transposed set)
        v2f A2 = *(const v2f*)(aw + 128);       // k2 + 2  (s-offset 4..7)
        v2f A3 = *(const v2f*)(aw + 160);
        v2f B0; B0.x = s0[0];  B0.y = s0[1];    // tile0, s = 2hh+{0,1}
        v2f B1; B1.x = s0[4];  B1.y = s0[5];    // tile0, s = 4+2hh+{0,1}
        v2f B2; B2.x = s0[16]; B2.y = s0[17];   // tile1
        v2f B3; B3.x = s0[20]; B3.y = s0[21];
        c00 = wmma4(A0, B0, c00);
        c01 = wmma4(A1, B0, c01);
        c00 = wmma4(A2, B1, c00);
        c01 = wmma4(A3, B1, c01);
        c10 = wmma4(A0, B2, c10);
        c11 = wmma4(A1, B2, c11);
        c10 = wmma4(A2, B3, c10);
        c11 = wmma4(A3, B3, c11);
        aw += 4 * 32 * 2;                        // next r => k2 advances by 4
      }
    }

#pragma unroll
  for (int rr = 0; rr < 8; ++rr) {
    int ch = rr + 8 * hh;
    float bb = (ch < 8) ? b0[ch] : b1[ch - 8];
    int base = ((((bi * 16 + ch) * XP + xo + 2) * XP + yo + 2) * XP + zo + 2) * WPAD + n + 2;
    float hv0 = fmaxf(c00[rr] + bb, 0.f) + fmaxf(c01[rr] + bb, 0.f);
    hpad[base] = hv0;                            // w = n (always < 25)
    if (16 + n < 25) {
      float hv1 = fmaxf(c10[rr] + bb, 0.f) + fmaxf(c11[rr] + bb, 0.f);
      hpad[base + 16] = hv1;
    }
  }
}

// ---------------- stage B: g = relu(conv2a(h) + b2a) (normal or transposed W) ----------------
// One wave = both w-tiles at one (b,x,y,z); A fragments shared (4 wmma per 6 b64 loads).

__global__ void __launch_bounds__(128)
convB_kernel(const float* __restrict__ src, const float* __restrict__ Aw,
             const float* __restrict__ bias, float* __restrict__ dst) {
  int wave = (blockIdx.x * blockDim.x + threadIdx.x) >> 5;
  int lane = threadIdx.x & 31;
  if (wave >= 2 * 25 * 25 * 25) return;
  int t = wave;
  int zo = t % 25; t /= 25;
  int yo = t % 25; t /= 25;
  int xo = t % 25;
  int bi = t / 25;

  const int n = lane & 15, hh = lane >> 4, m = n;
  v8f cA = {};   // w-tile 0
  v8f cB = {};   // w-tile 1

  for (int ci = 0; ci < 16; ++ci)
    for (int p = 0; p < 5; ++p)
      for (int q = 0; q < 5; ++q) {
        const float* srcq =
            src + ((((bi * 16 + ci) * XP + xo + p) * XP + yo + q) * XP + zo) * WPAD +
            n + 2 * hh;
        int base4 = (((ci * 5 + p) * 5 + q) * 5) * 4;
        const float* aw = Aw + (((base4 + hh) * 16) + m) * 2;
#pragma unroll
        for (int r = 0; r < 5; ++r) {
          const float* s0 = srcq + r * WPAD;
          v2f A0 = *(const v2f*)(aw);           // k2
          v2f A1 = *(const v2f*)(aw + 64);      // k2 + 2 (s-offset 4..7)
          v2f B0; B0.x = s0[0];  B0.y = s0[1];
          v2f B1; B1.x = s0[4];  B1.y = s0[5];
          v2f B2; B2.x = s0[16]; B2.y = s0[17];
          v2f B3; B3.x = s0[20]; B3.y = s0[21];
          cA = wmma4(A0, B0, cA);
          cA = wmma4(A1, B1, cA);
          cB = wmma4(A0, B2, cB);
          cB = wmma4(A1, B3, cB);
          aw += 4 * 16 * 2;
        }
      }

#pragma unroll
  for (int rr = 0; rr < 8; ++rr) {
    int co = rr + 8 * hh;
    float bb = bias[co];
    int base = ((((bi * 16 + co) * XP + xo + 2) * XP + yo + 2) * XP + zo + 2) * WPAD + n + 2;
    dst[base] = fmaxf(cA[rr] + bb, 0.f);
    if (16 + n < 25) dst[base + 16] = fmaxf(cB[rr] + bb, 0.f);
  }
}

// ---------------- conv2b (Co=1): out (=|+=) relu(conv2b(g) + b2b) ----------------

__global__ void conv2b_kernel(const float* __restrict__ gpad,
                              const float* __restrict__ w2b,
                              const float* __restrict__ b2b,
                              float* __restrict__ out, int transposed, int first) {
  int t = blockIdx.x * blockDim.x + threadIdx.x;
  if (t >= 2 * 25 * 25 * 25 * 25) return;
  int oidx = t;
  int wo = t % 25; t /= 25;
  int zo = t % 25; t /= 25;
  int yo = t % 25; t /= 25;
  int xo = t % 25;
  int bi = t / 25;

  float acc = b2b[0];
  for (int ci = 0; ci < 16; ++ci)
    for (int p = 0; p < 5; ++p)
      for (int q = 0; q < 5; ++q) {
        const float* srow =
            gpad + ((((bi * 16 + ci) * XP + xo + p) * XP + yo + q) * XP + zo) * WPAD + wo;
        for (int r = 0; r < 5; ++r) {
          const float* sv = srow + r * WPAD;
#pragma unroll
          for (int ss = 0; ss < 5; ++ss) {
            float wv = transposed
                           ? w2b[(((ci * 5 + r) * 5 + ss) * 5 + p) * 5 + q]
                           : w2b[(((ci * 5 + p) * 5 + q) * 5 + r) * 5 + ss];
            acc = fmaf(sv[ss], wv, acc);
          }
        }
      }
  float v = fmaxf(acc, 0.f);
  if (first) out[oidx] = v;
  else       out[oidx] += v;
}

// ---------------- launcher ----------------

extern "C" void kernel_launch(void* const* d_in, const int* in_sizes, int n_in,
                              void* d_out, int out_size, void* d_ws, size_t ws_size,
                              hipStream_t stream) {
  (void)in_sizes; (void)n_in; (void)out_size; (void)ws_size;
  const float* x   = (const float*)d_in[0];
  const float* w0  = (const float*)d_in[1];
  const float* b0  = (const float*)d_in[2];
  const float* w1  = (const float*)d_in[3];
  const float* b1  = (const float*)d_in[4];
  const float* w2a = (const float*)d_in[5];
  const float* b2a = (const float*)d_in[6];
  const float* w2b = (const float*)d_in[7];
  const float* b2b = (const float*)d_in[8];
  float* ws  = (float*)d_ws;
  float* out = (float*)d_out;

  float* xpad  = ws + OFF_XPAD;
  float* hpad  = ws + OFF_HPAD;
  float* gpad  = ws + OFF_GPAD;
  float* AwA   = ws + OFF_AWA;
  float* Aw2a  = ws + OFF_AW2A;
  float* Aw2aT = ws + OFF_AW2AT;

  // 1) zero pad buffers (x/h/g + slack), fully deterministic each call
  {
    int n4 = OFF_AWA / 4;
    zero_ws_kernel<<<(n4 + 255) / 256, 256, 0, stream>>>((float4*)ws, n4);
  }
  // 2) pack input + weights
  {
    int n = 2 * 25 * 25 * 25 * 25;
    pack_x_kernel<<<(n + 255) / 256, 256, 0, stream>>>(x, xpad);
    pack_wA_kernel<<<(SZ_AWA + 255) / 256, 256, 0, stream>>>(w0, w1, AwA);
    pack_w2a_kernel<<<(SZ_AW2A + 255) / 256, 256, 0, stream>>>(w2a, Aw2a, Aw2aT);
  }
  const int nWaves = 2 * 25 * 25 * 25;              // 31250 tiles (both w-tiles per wave)
  const int blocksT = (nWaves + 3) / 4;             // 4 waves / 128-thread block
  // 3) stage A -> hpad
  convA_kernel<<<blocksT, 128, 0, stream>>>(xpad, AwA, b0, b1, hpad);
  // 4) branch 1: conv2a -> gpad ; conv2b -> out
  convB_kernel<<<blocksT, 128, 0, stream>>>(hpad, Aw2a, b2a, gpad);
  {
    int n = 2 * 25 * 25 * 25 * 25;
    conv2b_kernel<<<(n + 255) / 256, 256, 0, stream>>>(gpad, w2b, b2b, out, 0, 1);
  }
  // 5) branch 2 (transposed weights): conv2aT -> gpad ; conv2bT += out
  convB_kernel<<<blocksT, 128, 0, stream>>>(hpad, Aw2aT, b2a, gpad);
  {
    int n = 2 * 25 * 25 * 25 * 25;
    conv2b_kernel<<<(n + 255) / 256, 256, 0, stream>>>(gpad, w2b, b2b, out, 1, 0);
  }
}